// ACF_model_16707422781605
// MI455X (gfx1250) — compile-verified
//
#include <hip/hip_runtime.h>

// ACF model, dead-code eliminated:
//   softmax over a size-1 axis == 1.0  =>  item attention weights are identically 1,
//   which also kills the entire component-attention / Fi path.
// Live path: row gathers + sum over L=32 Pi rows + one dot product per user.
// ~7.9 MB total traffic -> memory/launch bound on MI455X (23.3 TB/s).
// The L-reduction is done on the matrix pipe: V_WMMA_F32_16X16X4_F32 with B = ones
// (D[m,n] = sum_k A[m,k]), fp32-exact, same memory traffic as a VALU reduction.

#define NB    256   // batch
#define LPOS  32    // positives per user
#define NF    200   // FACTORS
#define NTILE 13    // ceil(200/16)

typedef __attribute__((ext_vector_type(2))) float v2f;
typedef __attribute__((ext_vector_type(8))) float v8f;

__global__ __launch_bounds__(64) void acf_live_kernel(
    const int*   __restrict__ user,
    const int*   __restrict__ item,
    const int*   __restrict__ user_pos,
    const float* __restrict__ Gu,
    const float* __restrict__ Gi,
    const float* __restrict__ Pi,
    float* __restrict__ out_xui,
    float* __restrict__ out_gu,
    float* __restrict__ out_gi,
    float* __restrict__ out_pi)
{
    const int u    = blockIdx.x;
    const int tid  = threadIdx.x;
    const int wave = tid >> 5;   // 2 waves per block (wave32)
    const int lane = tid & 31;

    __shared__ int   s_pos[LPOS];
    __shared__ float s_alla[NTILE * 16 + 16];
    __shared__ float s_red[64];

    if (tid < LPOS) s_pos[tid] = user_pos[u * LPOS + tid];
    __syncthreads();

    // ---- all_a[f] = sum_l Pi[pos[l], f] via V_WMMA_F32_16X16X4_F32 ----
    // A (16x4 f32) layout: lanes 0-15: M=lane, v[0]=K0,v[1]=K1 ; lanes 16-31: M=lane-16, v[0]=K2,v[1]=K3
    // B = all ones -> D[m,n] = sum_k A[m,k]  (B layout irrelevant)
    const int fm = lane & 15;          // feature within tile (M index)
    const int kb = (lane >> 4) * 2;    // K base for this lane half
    const v2f ones = {1.0f, 1.0f};

    for (int t = wave; t < NTILE; t += 2) {   // wave-uniform loop: EXEC all-ones at WMMA
        const int f  = t * 16 + fm;
        const int fc = (f < NF) ? f : 0;      // clamped address; value masked below
        v8f c = {0.f, 0.f, 0.f, 0.f, 0.f, 0.f, 0.f, 0.f};
        #pragma unroll
        for (int lc = 0; lc < 8; ++lc) {      // 8 K-chunks x K=4 = 32 positives
            const int  l0 = lc * 4 + kb;
            const long r0 = (long)s_pos[l0]     * NF;
            const long r1 = (long)s_pos[l0 + 1] * NF;
            const float p0 = Pi[r0 + fc];
            const float p1 = Pi[r1 + fc];
            v2f a;
            a[0] = (f < NF) ? p0 : 0.0f;      // branchless mask -> EXEC stays full
            a[1] = (f < NF) ? p1 : 0.0f;
            c = __builtin_amdgcn_wmma_f32_16x16x4_f32(
                    false, a, false, ones, (short)0, c, false, false);
        }
        // D layout: VGPR r -> M=r (lanes 0-15) / M=8+r (lanes 16-31); N = lane&15.
        // All N columns identical (B==ones): take N==0 lanes.
        if ((lane & 15) == 0) {
            const int mbase = (lane >> 4) * 8;
            #pragma unroll
            for (int r = 0; r < 8; ++r)
                s_alla[t * 16 + mbase + r] = c[r];
        }
    }
    __syncthreads();

    // ---- row copies + xui = dot(gu + all_a, gi) ----
    const long gub = (long)user[u] * NF;
    const long gib = (long)item[u] * NF;
    float part = 0.0f;
    for (int f = tid; f < NF; f += 64) {
        const float guv = Gu[gub + f];
        const float giv = Gi[gib + f];
        const float piv = Pi[gib + f];
        out_gu[u * NF + f] = guv;
        out_gi[u * NF + f] = giv;
        out_pi[u * NF + f] = piv;
        part += (guv + s_alla[f]) * giv;
    }
    s_red[tid] = part;
    __syncthreads();
    #pragma unroll
    for (int off = 32; off > 0; off >>= 1) {
        if (tid < off) s_red[tid] += s_red[tid + off];
        __syncthreads();
    }
    if (tid == 0) out_xui[u] = s_red[0];
}

extern "C" void kernel_launch(void* const* d_in, const int* in_sizes, int n_in,
                              void* d_out, int out_size, void* d_ws, size_t ws_size,
                              hipStream_t stream)
{
    const int*   user     = (const int*)  d_in[0];
    const int*   item     = (const int*)  d_in[1];
    const int*   user_pos = (const int*)  d_in[2];
    const float* Gu       = (const float*)d_in[3];
    const float* Gi       = (const float*)d_in[4];
    const float* Pi       = (const float*)d_in[5];
    // Weights d_in[6..18] are dead (softmax over singleton axis == 1).

    float* out = (float*)d_out;
    float* xui = out;                    // [256]
    float* ogu = xui + NB;               // [256*200]
    float* ogi = ogu + NB * NF;          // [256*200]
    float* opi = ogi + NB * NF;          // [256*200]

    acf_live_kernel<<<NB, 64, 0, stream>>>(user, item, user_pos, Gu, Gi, Pi,
                                           xui, ogu, ogi, opi);
}